// TransformerConv_14663018348976
// MI455X (gfx1250) — compile-verified
//
#include <hip/hip_runtime.h>
#include <hip/hip_bf16.h>
#include <math.h>

// GAT layer for MI455X (gfx1250), wave32.
// Strategy: fold W_src/attn_l and W_dst/attn_r into 512x8 vectors -> el/er is a
// skinny [N,512]x[512,16] GEMM; feat_v is the only full GEMM. Both use
// V_WMMA_F32_16X16X4_F32 (exact f32 math on the matrix pipe). Edge softmax via
// atomic-max/exp/atomic-add; aggregation is wave-per-edge gather + f32 atomic
// scatter into the L2-resident output.

#define Nn 50000
#define Ee 400000
#define INF_ 512
#define HF_ 512
#define HEADS_ 8
#define OUTF_ 64
#define NEG_SLOPE_ 0.2f

typedef __attribute__((ext_vector_type(2))) float v2f;
typedef __attribute__((ext_vector_type(8))) float v8f;

// ---------------------------------------------------------------- fold weights
// P[k][0..7]  = sum_f W_src[k, h*64+f] * attn_l[h,f]
// P[k][8..15] = sum_f W_dst[k, h*64+f] * attn_r[h,f]
// cbias[0..7] = b_src . attn_l per head ; cbias[8..15] = b_dst . attn_r
__global__ void gat_fold(const float* __restrict__ Wsrc, const float* __restrict__ Wdst,
                         const float* __restrict__ attn_l, const float* __restrict__ attn_r,
                         const float* __restrict__ bsrc, const float* __restrict__ bdst,
                         float* __restrict__ P, float* __restrict__ cbias) {
  int idx = blockIdx.x * blockDim.x + threadIdx.x;
  if (idx < INF_ * 16) {
    int k = idx >> 4, col = idx & 15, h = col & 7;
    const float* W  = (col < 8) ? Wsrc : Wdst;
    const float* av = (col < 8) ? attn_l : attn_r;
    float s = 0.f;
#pragma unroll 4
    for (int f = 0; f < OUTF_; ++f)
      s += W[(size_t)k * HF_ + h * OUTF_ + f] * av[h * OUTF_ + f];
    P[k * 16 + col] = s;
  }
  if (idx < 16) {
    int col = idx, h = col & 7;
    const float* b  = (col < 8) ? bsrc : bdst;
    const float* av = (col < 8) ? attn_l : attn_r;
    float s = 0.f;
    for (int f = 0; f < OUTF_; ++f) s += b[h * OUTF_ + f] * av[h * OUTF_ + f];
    cbias[col] = s;
  }
}

// ---------------------------------------------- el|er skinny GEMM: [N,512]x[512,16]
// One wave per 16-row tile; cols 0..7 = el heads, 8..15 = er heads.
__global__ __launch_bounds__(256) void gat_eler_gemm(const float* __restrict__ feat,
                                                     const float* __restrict__ P,
                                                     const float* __restrict__ cbias,
                                                     float* __restrict__ eler) {
  const int tile = blockIdx.x * 8 + (threadIdx.x >> 5);
  if (tile >= Nn / 16) return;                  // whole-wave exit: EXEC stays all-1s
  const int lane = threadIdx.x & 31;
  const int row0 = tile * 16;
  const int m = lane & 15;
  const int hi = lane >> 4;
  const int kb = hi * 2;
  const int n = lane & 15;

  v8f acc = {};
  for (int k0 = 0; k0 < INF_; k0 += 4) {
    const float2 av = *(const float2*)(feat + (size_t)(row0 + m) * INF_ + k0 + kb);
    v2f a; a.x = av.x; a.y = av.y;
    v2f b; b.x = P[(k0 + kb) * 16 + n];
           b.y = P[(k0 + kb + 1) * 16 + n];
    acc = __builtin_amdgcn_wmma_f32_16x16x4_f32(false, a, false, b, (short)0, acc, false, false);
  }
  const float cb = cbias[n];
#pragma unroll
  for (int r = 0; r < 8; ++r) {
    const int mr = r + 8 * hi;
    eler[(size_t)(row0 + mr) * 16 + n] = acc[r] + cb;
  }
}

// ---------------------------------------------- feat_v GEMM: [N,512]x[512,512]+b
// Block = 8 waves: 16-row x 512-col tile. A tile staged in LDS (stride 516 to
// spread banks); each wave owns 64 cols with 4 v8f accumulators.
__global__ __launch_bounds__(256) void gat_featv_gemm(const float* __restrict__ feat,
                                                      const float* __restrict__ Wv,
                                                      const float* __restrict__ bv,
                                                      float* __restrict__ featv) {
  __shared__ float ldsA[16 * 516];
  const int tid = threadIdx.x;
  const int wave = tid >> 5;
  const int lane = tid & 31;
  const int row0 = blockIdx.x * 16;

  // cooperative 16x512 f32 tile load, float4 per thread x8
#pragma unroll
  for (int it = 0; it < 8; ++it) {
    int idx = it * 256 + tid;
    int r = idx >> 7;
    int c4 = idx & 127;
    const float4 v = *(const float4*)(feat + (size_t)(row0 + r) * INF_ + c4 * 4);
    *(float4*)(&ldsA[r * 516 + c4 * 4]) = v;
  }
  __syncthreads();

  const int m = lane & 15;
  const int hi = lane >> 4;
  const int kb = hi * 2;
  const int n = lane & 15;
  const int col0 = wave * 64;

  v8f acc[4] = {};
  for (int k0 = 0; k0 < INF_; k0 += 4) {
    const float2 av = *(const float2*)(&ldsA[m * 516 + k0 + kb]);
    v2f a; a.x = av.x; a.y = av.y;
#pragma unroll
    for (int j = 0; j < 4; ++j) {
      const int c = col0 + j * 16 + n;
      v2f b; b.x = Wv[(size_t)(k0 + kb) * HF_ + c];
             b.y = Wv[(size_t)(k0 + kb + 1) * HF_ + c];
      acc[j] = __builtin_amdgcn_wmma_f32_16x16x4_f32(false, a, false, b, (short)0, acc[j], false, false);
    }
  }
#pragma unroll
  for (int j = 0; j < 4; ++j) {
    const int c = col0 + j * 16 + n;
    const float bias = bv[c];
#pragma unroll
    for (int r = 0; r < 8; ++r) {
      const int mr = r + 8 * hi;
      featv[(size_t)(row0 + mr) * HF_ + c] = acc[j][r] + bias;
    }
  }
}

// ---------------------------------------------------------------- init buffers
__global__ void gat_init(float* __restrict__ out, int* __restrict__ emax,
                         float* __restrict__ denom, int nOut, int nNH) {
  int i = blockIdx.x * blockDim.x + threadIdx.x;
  if (i < nOut) out[i] = 0.f;
  if (i < nNH) { emax[i] = (int)0xFF800000; denom[i] = 0.f; }  // -inf
}

__device__ inline void atomicMaxF32(float* addr, float val) {
  // monotone int trick (valid for all finite floats incl. mixed signs, -inf init)
  if (val >= 0.f) atomicMax((int*)addr, __float_as_int(val));
  else            atomicMin((unsigned int*)addr, __float_as_uint(val));
}

__device__ inline float edge_logit(const float* eler, int s, int d, int h) {
  float x = eler[(size_t)s * 16 + h] + eler[(size_t)d * 16 + 8 + h];
  return (x >= 0.f) ? x : NEG_SLOPE_ * x;
}

// ------------------------------------------------------- pass 1: segment max
__global__ void gat_edge_max(const int* __restrict__ src, const int* __restrict__ dst,
                             const float* __restrict__ eler, float* __restrict__ emax) {
  int i = blockIdx.x * blockDim.x + threadIdx.x;
  if (i >= Ee * HEADS_) return;
  int e = i >> 3, h = i & 7;
  int s = src[e], d = dst[e];
  atomicMaxF32(&emax[(size_t)d * HEADS_ + h], edge_logit(eler, s, d, h));
}

// --------------------------------------------- pass 2: exp + segment sum denom
__global__ void gat_edge_exp(const int* __restrict__ src, const int* __restrict__ dst,
                             const float* __restrict__ eler, const float* __restrict__ emax,
                             float* __restrict__ expbuf, float* __restrict__ denom) {
  int i = blockIdx.x * blockDim.x + threadIdx.x;
  if (i >= Ee * HEADS_) return;
  int e = i >> 3, h = i & 7;
  int s = src[e], d = dst[e];
  float x = edge_logit(eler, s, d, h);
  float ex = __expf(x - emax[(size_t)d * HEADS_ + h]);
  expbuf[(size_t)e * HEADS_ + h] = ex;
  atomicAdd(&denom[(size_t)d * HEADS_ + h], ex);
}

// ----------------------------------------- pass 3: gather * a -> atomic scatter
// One wave per edge; lanes 0..7 compute a[h], shfl-broadcast, stream 512 floats.
__global__ __launch_bounds__(256) void gat_aggregate(const int* __restrict__ src,
                                                     const int* __restrict__ dst,
                                                     const float* __restrict__ expbuf,
                                                     const float* __restrict__ denom,
                                                     const float* __restrict__ featv,
                                                     float* __restrict__ out) {
  const int e = blockIdx.x * 8 + (threadIdx.x >> 5);
  if (e >= Ee) return;                          // whole-wave exit
  const int lane = threadIdx.x & 31;
  const int s = src[e], d = dst[e];
  float ah = 0.f;
  if (lane < HEADS_) {
    float dn = denom[(size_t)d * HEADS_ + lane];
    ah = expbuf[(size_t)e * HEADS_ + lane] / fmaxf(dn, 1e-16f);
  }
  const float* vs = featv + (size_t)s * HF_;
  float* od = out + (size_t)d * HF_;
#pragma unroll
  for (int it = 0; it < 4; ++it) {
    const int c4 = lane + it * 32;              // float4 chunk 0..127
    const int h = c4 >> 4;                      // 16 chunks per head
    const float a = __shfl(ah, h, 32);
    const float4 v = *(const float4*)(vs + c4 * 4);
    atomicAdd(&od[c4 * 4 + 0], v.x * a);
    atomicAdd(&od[c4 * 4 + 1], v.y * a);
    atomicAdd(&od[c4 * 4 + 2], v.z * a);
    atomicAdd(&od[c4 * 4 + 3], v.w * a);
  }
}

extern "C" void kernel_launch(void* const* d_in, const int* in_sizes, int n_in,
                              void* d_out, int out_size, void* d_ws, size_t ws_size,
                              hipStream_t stream) {
  const float* feat   = (const float*)d_in[0];
  const int*   src    = (const int*)d_in[1];
  const int*   dst    = (const int*)d_in[2];
  const float* W_src  = (const float*)d_in[3];
  const float* b_src  = (const float*)d_in[4];
  const float* W_dst  = (const float*)d_in[5];
  const float* b_dst  = (const float*)d_in[6];
  const float* W_val  = (const float*)d_in[7];
  const float* b_val  = (const float*)d_in[8];
  const float* attn_l = (const float*)d_in[9];
  const float* attn_r = (const float*)d_in[10];
  float* out = (float*)d_out;

  // workspace layout (floats): ~122 MB total
  float* ws     = (float*)d_ws;
  float* featv  = ws;                                   // N*512
  float* eler   = featv  + (size_t)Nn * HF_;            // N*16
  float* P      = eler   + (size_t)Nn * 16;             // 512*16
  float* cbias  = P      + (size_t)INF_ * 16;           // 16
  float* emax   = cbias  + 16;                          // N*8
  float* denom  = emax   + (size_t)Nn * HEADS_;         // N*8
  float* expbuf = denom  + (size_t)Nn * HEADS_;         // E*8

  gat_fold<<<(INF_ * 16 + 255) / 256, 256, 0, stream>>>(W_src, W_dst, attn_l, attn_r,
                                                        b_src, b_dst, P, cbias);
  gat_eler_gemm<<<(Nn / 16 + 7) / 8, 256, 0, stream>>>(feat, P, cbias, eler);
  gat_featv_gemm<<<Nn / 16, 256, 0, stream>>>(feat, W_val, b_val, featv);

  const int nOut = Nn * HF_;
  gat_init<<<(nOut + 255) / 256, 256, 0, stream>>>(out, (int*)emax, denom, nOut, Nn * HEADS_);
  gat_edge_max<<<(Ee * HEADS_ + 255) / 256, 256, 0, stream>>>(src, dst, eler, emax);
  gat_edge_exp<<<(Ee * HEADS_ + 255) / 256, 256, 0, stream>>>(src, dst, eler, emax, expbuf, denom);
  gat_aggregate<<<Ee / 8, 256, 0, stream>>>(src, dst, expbuf, denom, featv, out);
}